// RigidBodySimulator_36378372997392
// MI455X (gfx1250) — compile-verified
//
#include <hip/hip_runtime.h>
#include <stdint.h>

// ---------------- problem constants (from the reference) ----------------
#define NBLK    256                      // persistent workgroups (must all be co-resident)
#define TPB     256                      // 8 waves of 32
#define NVERTS  2000000
#define NGROUP  (NVERTS/4)               // 500000 groups of 4 vertices
#define GPB     ((NGROUP + NBLK - 1)/NBLK)   // 1954 groups per block (max)
#define TSTEPS  240                      // FRAMES*SUBSTEP
#define DT_F    (1.0f/600.0f)
#define COS_S_F 0.93969262f              // cos(20 deg)
#define SIN_S_F 0.34202014f              // sin(20 deg)
#define NX_F    (-SIN_S_F)               // CONTACT_N.x
#define NY_F    (COS_S_F)                // CONTACT_N.y  (z component is 0)
#define INVM_F  (1.0f/2000000.0f)

// dynamic LDS layout:
//   [0 .. GPB*3*16)          float4 xbuf   (this block's vertex slice, ~91.6 KB)
//   [.. +16*16)              float4 red    (8 wave partials x 2 uses)
//   [.. +64)                 float  par[16] (per-substep mask params)
#define XBUF_BYTES ((size_t)GPB*3*sizeof(float4))
#define SMEM_BYTES (XBUF_BYTES + 16*sizeof(float4) + 16*sizeof(float))

// ---------------- small 3x3 helpers (scalar, thread 0 only) ----------------
__device__ __forceinline__ void mm3(float* o, const float* a, const float* b) {
#pragma unroll
  for (int i = 0; i < 3; ++i)
#pragma unroll
    for (int j = 0; j < 3; ++j)
      o[i*3+j] = a[i*3+0]*b[0*3+j] + a[i*3+1]*b[1*3+j] + a[i*3+2]*b[2*3+j];
}

__device__ __forceinline__ void inv3(const float* m, float* o) {
  float c00 = m[4]*m[8] - m[5]*m[7];
  float c01 = m[5]*m[6] - m[3]*m[8];
  float c02 = m[3]*m[7] - m[4]*m[6];
  float det = m[0]*c00 + m[1]*c01 + m[2]*c02;
  float id  = 1.0f/det;
  o[0]=c00*id;                 o[1]=(m[2]*m[7]-m[1]*m[8])*id; o[2]=(m[1]*m[5]-m[2]*m[4])*id;
  o[3]=c01*id;                 o[4]=(m[0]*m[8]-m[2]*m[6])*id; o[5]=(m[2]*m[3]-m[0]*m[5])*id;
  o[6]=c02*id;                 o[7]=(m[1]*m[6]-m[0]*m[7])*id; o[8]=(m[0]*m[4]-m[1]*m[3])*id;
}

// ---------------- persistent simulation kernel ----------------
__global__ __launch_bounds__(TPB, 1)
void rb_sim_kernel(const float4* __restrict__ xg,
                   const float* __restrict__ mcp,
                   const float* __restrict__ itp,
                   const float* __restrict__ iqp,
                   const float* __restrict__ ivp,
                   const float* __restrict__ knp,
                   const float* __restrict__ mup,
                   const float* __restrict__ ldp,
                   const float* __restrict__ adp,
                   float* __restrict__ out,
                   unsigned* __restrict__ subSlots,   // 2*NBLK slots of 8 u32 (ping-pong)
                   unsigned* __restrict__ inSlots)    // NBLK   slots of 8 u32 (inertia)
{
  extern __shared__ char smem_raw[];
  float4* xbuf = (float4*)smem_raw;
  float4* red  = (float4*)(smem_raw + XBUF_BYTES);                 // 16 entries
  float*  par  = (float*)(smem_raw + XBUF_BYTES + 16*sizeof(float4));

  const int tid  = threadIdx.x;
  const int blk  = blockIdx.x;
  const int wave = tid >> 5;
  const int lane = tid & 31;
  const int g0   = blk * GPB;
  int gcnt = NGROUP - g0; if (gcnt > GPB) gcnt = GPB;
  const int nf4  = gcnt * 3;

  // ---- Phase A: async-copy this block's vertex slice into LDS (done ONCE) ----
  {
    const float4* src = xg + (size_t)g0 * 3;
    for (int i = tid; i < nf4; i += TPB) {
      unsigned lds = (unsigned)(size_t)(&xbuf[i]);               // low 32 bits of generic ptr = LDS offset
      unsigned long long ga = (unsigned long long)(size_t)(src + i);
      asm volatile("global_load_async_to_lds_b128 %0, %1, off"
                   :: "v"(lds), "v"(ga) : "memory");
    }
    asm volatile("s_wait_asynccnt 0" ::: "memory");
  }
  __syncthreads();

  const float mcx = mcp[0], mcy = mcp[1], mcz = mcp[2];

  // ---- Phase B: inertia = sum(r.r)I - sum(outer(r,r)) over all vertices ----
  float axx=0.f, ayy=0.f, azz=0.f, axy=0.f, axz=0.f, ayz=0.f;
  for (int g = tid; g < gcnt; g += TPB) {
    float4 f0 = xbuf[3*g+0], f1 = xbuf[3*g+1], f2 = xbuf[3*g+2];
#define IACC(X,Y,Z) do { float rx=(X)-mcx, ry=(Y)-mcy, rz=(Z)-mcz; \
    axx+=rx*rx; ayy+=ry*ry; azz+=rz*rz; axy+=rx*ry; axz+=rx*rz; ayz+=ry*rz; } while(0)
    IACC(f0.x, f0.y, f0.z);
    IACC(f0.w, f1.x, f1.y);
    IACC(f1.z, f1.w, f2.x);
    IACC(f2.y, f2.z, f2.w);
#undef IACC
  }
  // wave-level shuffle reduce (deterministic butterfly)
#pragma unroll
  for (int off = 16; off > 0; off >>= 1) {
    axx += __shfl_down(axx, off, 32);
    ayy += __shfl_down(ayy, off, 32);
    azz += __shfl_down(azz, off, 32);
    axy += __shfl_down(axy, off, 32);
    axz += __shfl_down(axz, off, 32);
    ayz += __shfl_down(ayz, off, 32);
  }
  if (lane == 0) {
    float4 v1; v1.x=axx; v1.y=ayy; v1.z=azz; v1.w=axy;
    float4 v2; v2.x=axz; v2.y=ayz; v2.z=0.f; v2.w=0.f;
    red[wave]   = v1;
    red[8+wave] = v2;
  }
  __syncthreads();

  // wave 0: combine 8 wave partials, publish, arrive+reduce across blocks
  float t_xx=0.f, t_yy=0.f, t_zz=0.f, t_xy=0.f, t_xz=0.f, t_yz=0.f;
  if (tid < 32) {
    float4 v1 = {0.f,0.f,0.f,0.f}, v2 = {0.f,0.f,0.f,0.f};
    if (tid < 8) { v1 = red[tid]; v2 = red[8+tid]; }
    float bxx=v1.x, byy=v1.y, bzz=v1.z, bxy=v1.w, bxz=v2.x, byz=v2.y;
#pragma unroll
    for (int off = 4; off > 0; off >>= 1) {
      bxx += __shfl_down(bxx, off, 8);
      byy += __shfl_down(byy, off, 8);
      bzz += __shfl_down(bzz, off, 8);
      bxy += __shfl_down(bxy, off, 8);
      bxz += __shfl_down(bxz, off, 8);
      byz += __shfl_down(byz, off, 8);
    }
    if (tid == 0) {
      unsigned* s = inSlots + blk*8;
      __hip_atomic_store(&s[0], __float_as_uint(bxx), __ATOMIC_RELAXED, __HIP_MEMORY_SCOPE_AGENT);
      __hip_atomic_store(&s[1], __float_as_uint(byy), __ATOMIC_RELAXED, __HIP_MEMORY_SCOPE_AGENT);
      __hip_atomic_store(&s[2], __float_as_uint(bzz), __ATOMIC_RELAXED, __HIP_MEMORY_SCOPE_AGENT);
      __hip_atomic_store(&s[3], __float_as_uint(bxy), __ATOMIC_RELAXED, __HIP_MEMORY_SCOPE_AGENT);
      __hip_atomic_store(&s[4], __float_as_uint(bxz), __ATOMIC_RELAXED, __HIP_MEMORY_SCOPE_AGENT);
      __hip_atomic_store(&s[5], __float_as_uint(byz), __ATOMIC_RELAXED, __HIP_MEMORY_SCOPE_AGENT);
      __hip_atomic_store(&s[7], 1u, __ATOMIC_RELEASE, __HIP_MEMORY_SCOPE_AGENT);
    }
    for (int bb = tid; bb < NBLK; bb += 32) {
      unsigned* s = inSlots + bb*8;
      while (__hip_atomic_load(&s[7], __ATOMIC_ACQUIRE, __HIP_MEMORY_SCOPE_AGENT) < 1u)
        __builtin_amdgcn_s_sleep(1);
      t_xx += __uint_as_float(__hip_atomic_load(&s[0], __ATOMIC_RELAXED, __HIP_MEMORY_SCOPE_AGENT));
      t_yy += __uint_as_float(__hip_atomic_load(&s[1], __ATOMIC_RELAXED, __HIP_MEMORY_SCOPE_AGENT));
      t_zz += __uint_as_float(__hip_atomic_load(&s[2], __ATOMIC_RELAXED, __HIP_MEMORY_SCOPE_AGENT));
      t_xy += __uint_as_float(__hip_atomic_load(&s[3], __ATOMIC_RELAXED, __HIP_MEMORY_SCOPE_AGENT));
      t_xz += __uint_as_float(__hip_atomic_load(&s[4], __ATOMIC_RELAXED, __HIP_MEMORY_SCOPE_AGENT));
      t_yz += __uint_as_float(__hip_atomic_load(&s[5], __ATOMIC_RELAXED, __HIP_MEMORY_SCOPE_AGENT));
    }
#pragma unroll
    for (int off = 16; off > 0; off >>= 1) {
      t_xx += __shfl_down(t_xx, off, 32);
      t_yy += __shfl_down(t_yy, off, 32);
      t_zz += __shfl_down(t_zz, off, 32);
      t_xy += __shfl_down(t_xy, off, 32);
      t_xz += __shfl_down(t_xz, off, 32);
      t_yz += __shfl_down(t_yz, off, 32);
    }
  }

  // ---- thread-0 persistent state (every block computes identical values) ----
  float in00=0.f,in01=0.f,in02=0.f,in11=0.f,in12=0.f,in22=0.f;
  float tx=0.f,ty=0.f,tz=0.f, qw=1.f,qx=0.f,qy=0.f,qz=0.f;
  float vx=0.f,vy=0.f,vz=0.f, wxo=0.f,wyo=0.f,wzo=0.f;
  float knv=0.f,muv=0.f,ldv=0.f,adv=0.f;
  float R00=1.f,R01=0.f,R02=0.f,R10=0.f,R11=1.f,R12=0.f,R20=0.f,R21=0.f,R22=1.f;
  if (tid == 0) {
    in00 = t_yy + t_zz; in11 = t_xx + t_zz; in22 = t_xx + t_yy;
    in01 = -t_xy; in02 = -t_xz; in12 = -t_yz;
    tx = itp[0]; ty = itp[1]; tz = itp[2];
    float q0=iqp[0], q1=iqp[1], q2=iqp[2], q3=iqp[3];
    float qn = 1.0f/sqrtf(q0*q0+q1*q1+q2*q2+q3*q3);
    qw=q0*qn; qx=q1*qn; qy=q2*qn; qz=q3*qn;
    vx=ivp[0]; vy=ivp[1]; vz=ivp[2];
    knv=knp[0]; muv=mup[0]; ldv=ldp[0]; adv=adp[0];
  }

  // ---- Phase C: 240 substeps, all vertex traffic from LDS ----
  for (int t = 0; t < TSTEPS; ++t) {
    if (tid == 0) {
      float qn = 1.0f/sqrtf(qw*qw+qx*qx+qy*qy+qz*qz);
      float w=qw*qn, X=qx*qn, Y=qy*qn, Z=qz*qn;
      R00 = 1.0f - 2.0f*(Y*Y + Z*Z);  R01 = 2.0f*(X*Y - w*Z);  R02 = 2.0f*(X*Z + w*Y);
      R10 = 2.0f*(X*Y + w*Z);  R11 = 1.0f - 2.0f*(X*X + Z*Z);  R12 = 2.0f*(Y*Z - w*X);
      R20 = 2.0f*(X*Z - w*Y);  R21 = 2.0f*(Y*Z + w*X);  R22 = 1.0f - 2.0f*(X*X + Y*Y);
      // a = R^T n ;  mask1: x.a < th1
      float pax = R00*NX_F + R10*NY_F;
      float pay = R01*NX_F + R11*NY_F;
      float paz = R02*NX_F + R12*NY_F;
      float s1  = (tx+mcx)*NX_F + (ty+mcy)*NY_F;        // n.z == 0
      float th1 = -COS_S_F*0.1f - s1;
      // b = R^T (n x omega) ; mask2: x.b < th2
      float cwx = NY_F*wzo;
      float cwy = -NX_F*wzo;
      float cwz = NX_F*wyo - NY_F*wxo;
      float pbx = R00*cwx + R10*cwy + R20*cwz;
      float pby = R01*cwx + R11*cwy + R21*cwz;
      float pbz = R02*cwx + R12*cwy + R22*cwz;
      float th2 = -(vx*NX_F + vy*NY_F);
      par[0]=pax; par[1]=pay; par[2]=paz; par[3]=th1;
      par[4]=pbx; par[5]=pby; par[6]=pbz; par[7]=th2;
    }
    __syncthreads();                                      // (1) params visible
    const float ax=par[0], ay=par[1], az=par[2], th1=par[3];
    const float bx=par[4], by=par[5], bz=par[6], th2=par[7];

    float c=0.f, sx=0.f, sy=0.f, sz=0.f;
#define MACC(X,Y,Z) do { float Xv=(X), Yv=(Y), Zv=(Z); \
    float da = ax*Xv + ay*Yv + az*Zv; \
    float db = bx*Xv + by*Yv + bz*Zv; \
    float m  = ((da < th1) && (db < th2)) ? 1.0f : 0.0f; \
    c += m; sx += m*Xv; sy += m*Yv; sz += m*Zv; } while(0)
#define GBODY(G) do { float4 f0 = xbuf[3*(G)+0], f1 = xbuf[3*(G)+1], f2 = xbuf[3*(G)+2]; \
    MACC(f0.x, f0.y, f0.z); \
    MACC(f0.w, f1.x, f1.y); \
    MACC(f1.z, f1.w, f2.x); \
    MACC(f2.y, f2.z, f2.w); } while(0)
    int g = tid;
    for (; g + TPB < gcnt; g += 2*TPB) { GBODY(g); GBODY(g + TPB); }
    for (; g < gcnt; g += TPB)         { GBODY(g); }
#undef GBODY
#undef MACC

    // wave-level shuffle reduce (no barriers)
#pragma unroll
    for (int off = 16; off > 0; off >>= 1) {
      c  += __shfl_down(c,  off, 32);
      sx += __shfl_down(sx, off, 32);
      sy += __shfl_down(sy, off, 32);
      sz += __shfl_down(sz, off, 32);
    }
    if (lane == 0) { float4 v; v.x=c; v.y=sx; v.z=sy; v.w=sz; red[wave] = v; }
    __syncthreads();                                      // (2) wave partials visible

    const int parity = t & 1;
    float num=0.f, gsx=0.f, gsy=0.f, gsz=0.f;
    if (tid < 32) {
      // combine the 8 wave partials (width-8 shuffle)
      float4 v = {0.f,0.f,0.f,0.f};
      if (tid < 8) v = red[tid];
      float bc=v.x, bsx=v.y, bsy=v.z, bsz=v.w;
#pragma unroll
      for (int off = 4; off > 0; off >>= 1) {
        bc  += __shfl_down(bc,  off, 8);
        bsx += __shfl_down(bsx, off, 8);
        bsy += __shfl_down(bsy, off, 8);
        bsz += __shfl_down(bsz, off, 8);
      }
      if (tid == 0) {
        unsigned* s = subSlots + (size_t)(parity*NBLK + blk)*8;
        __hip_atomic_store(&s[0], __float_as_uint(bc),  __ATOMIC_RELAXED, __HIP_MEMORY_SCOPE_AGENT);
        __hip_atomic_store(&s[1], __float_as_uint(bsx), __ATOMIC_RELAXED, __HIP_MEMORY_SCOPE_AGENT);
        __hip_atomic_store(&s[2], __float_as_uint(bsy), __ATOMIC_RELAXED, __HIP_MEMORY_SCOPE_AGENT);
        __hip_atomic_store(&s[3], __float_as_uint(bsz), __ATOMIC_RELAXED, __HIP_MEMORY_SCOPE_AGENT);
        __hip_atomic_store(&s[7], (unsigned)(t+1), __ATOMIC_RELEASE, __HIP_MEMORY_SCOPE_AGENT);
      }
      // fused grid "barrier" + deterministic cross-block reduce
      const unsigned want = (unsigned)(t + 1);
      for (int bb = tid; bb < NBLK; bb += 32) {
        unsigned* s = subSlots + (size_t)(parity*NBLK + bb)*8;
        while (__hip_atomic_load(&s[7], __ATOMIC_ACQUIRE, __HIP_MEMORY_SCOPE_AGENT) < want)
          __builtin_amdgcn_s_sleep(1);
        num += __uint_as_float(__hip_atomic_load(&s[0], __ATOMIC_RELAXED, __HIP_MEMORY_SCOPE_AGENT));
        gsx += __uint_as_float(__hip_atomic_load(&s[1], __ATOMIC_RELAXED, __HIP_MEMORY_SCOPE_AGENT));
        gsy += __uint_as_float(__hip_atomic_load(&s[2], __ATOMIC_RELAXED, __HIP_MEMORY_SCOPE_AGENT));
        gsz += __uint_as_float(__hip_atomic_load(&s[3], __ATOMIC_RELAXED, __HIP_MEMORY_SCOPE_AGENT));
      }
#pragma unroll
      for (int off = 16; off > 0; off >>= 1) {
        num += __shfl_down(num, off, 32);
        gsx += __shfl_down(gsx, off, 32);
        gsy += __shfl_down(gsy, off, 32);
        gsz += __shfl_down(gsz, off, 32);
      }
    }

    if (tid == 0) {
      // ---- finalize (identical in every block; deterministic) ----
      float numf = fmaxf(num, 1.0f);
      float rix = gsx/numf, riy = gsy/numf, riz = gsz/numf;
      float Rix = R00*rix + R01*riy + R02*riz;
      float Riy = R10*rix + R11*riy + R12*riz;
      float Riz = R20*rix + R21*riy + R22*riz;
      float vix = vx + (wyo*Riz - wzo*Riy);
      float viy = vy + (wzo*Rix - wxo*Riz);
      float viz = vz + (wxo*Riy - wyo*Rix);
      float vin = vix*NX_F + viy*NY_F;
      float vnx = vin*NX_F, vny = vin*NY_F;
      float vtx = vix - vnx, vty = viy - vny, vtz = viz;
      float nvn = fabsf(vin);
      float nvt = sqrtf(vtx*vtx + vty*vty + vtz*vtz);
      float alpha = fmaxf(1.0f - muv*(1.0f+knv)*(nvn/(nvt + 1e-6f)), 0.0f);
      float vnewx = -knv*vnx + alpha*vtx;
      float vnewy = -knv*vny + alpha*vty;
      float vnewz = alpha*vtz;
      // I = R * inertia * R^T ; I_inv ; kmat ; J
      float Rm[9] = {R00,R01,R02, R10,R11,R12, R20,R21,R22};
      float In[9] = {in00,in01,in02, in01,in11,in12, in02,in12,in22};
      float T[9]; mm3(T, Rm, In);
      float Iw[9];
#pragma unroll
      for (int i = 0; i < 3; ++i)
#pragma unroll
        for (int j = 0; j < 3; ++j)
          Iw[i*3+j] = T[i*3+0]*Rm[j*3+0] + T[i*3+1]*Rm[j*3+1] + T[i*3+2]*Rm[j*3+2];
      float Ii[9]; inv3(Iw, Ii);
      float C[9]  = {0.0f,-Riz,Riy, Riz,0.0f,-Rix, -Riy,Rix,0.0f};
      float A1[9]; mm3(A1, Ii, C);     // I_inv @ Rri
      float B1[9]; mm3(B1, C, A1);     // Rri @ I_inv @ Rri
      float K[9]  = { INVM_F - B1[0], -B1[1],          -B1[2],
                      -B1[3],          INVM_F - B1[4], -B1[5],
                      -B1[6],          -B1[7],          INVM_F - B1[8] };
      float Ki[9]; inv3(K, Ki);
      float rhx = vnewx - vix, rhy = vnewy - viy, rhz = vnewz - viz;
      float Jx = Ki[0]*rhx + Ki[1]*rhy + Ki[2]*rhz;
      float Jy = Ki[3]*rhx + Ki[4]*rhy + Ki[5]*rhz;
      float Jz = Ki[6]*rhx + Ki[7]*rhy + Ki[8]*rhz;
      float hitf = (num > 0.0f) ? 1.0f : 0.0f;
      float dvx = hitf * Jx * INVM_F;
      float dvy = hitf * Jy * INVM_F;
      float dvz = hitf * Jz * INVM_F;
      float dwx = hitf * (A1[0]*Jx + A1[1]*Jy + A1[2]*Jz);
      float dwy = hitf * (A1[3]*Jx + A1[4]*Jy + A1[5]*Jz);
      float dwz = hitf * (A1[6]*Jx + A1[7]*Jy + A1[8]*Jz);
      vx = vx*ldv + dvx;
      vy = (vy - 9.8f*DT_F)*ldv + dvy;
      vz = vz*ldv + dvz;
      wxo = wxo*adv + dwx;
      wyo = wyo*adv + dwy;
      wzo = wzo*adv + dwz;
      tx += DT_F*vx; ty += DT_F*vy; tz += DT_F*vz;
      float hx = wxo*(0.5f*DT_F), hy = wyo*(0.5f*DT_F), hz = wzo*(0.5f*DT_F);
      float dqw = -(hx*qx + hy*qy + hz*qz);
      float dqx =  hx*qw + hy*qz - hz*qy;
      float dqy =  hy*qw + hz*qx - hx*qz;
      float dqz =  hz*qw + hx*qy - hy*qx;
      qw += dqw; qx += dqx; qy += dqy; qz += dqz;
      float qni = 1.0f/sqrtf(qw*qw+qx*qx+qy*qy+qz*qz);
      qw*=qni; qx*=qni; qy*=qni; qz*=qni;
      if (blk == 0) {
        float* o = out + (size_t)t*7;
        o[0]=tx; o[1]=ty; o[2]=tz; o[3]=qw; o[4]=qx; o[5]=qy; o[6]=qz;
      }
    }
    // other threads resynchronize at the top-of-loop __syncthreads()
  }
}

// ---------------- host launcher ----------------
extern "C" void kernel_launch(void* const* d_in, const int* in_sizes, int n_in,
                              void* d_out, int out_size, void* d_ws, size_t ws_size,
                              hipStream_t stream) {
  (void)in_sizes; (void)n_in; (void)out_size; (void)ws_size;
  const float4* x  = (const float4*)d_in[0];
  const float*  mc = (const float*)d_in[1];
  const float*  it = (const float*)d_in[2];
  const float*  iq = (const float*)d_in[3];
  const float*  iv = (const float*)d_in[4];
  const float*  kn = (const float*)d_in[5];
  const float*  mu = (const float*)d_in[6];
  const float*  ld = (const float*)d_in[7];
  const float*  ad = (const float*)d_in[8];
  float* out = (float*)d_out;

  unsigned* subSlots = (unsigned*)d_ws;                       // 2*NBLK*8 u32
  unsigned* inSlots  = (unsigned*)d_ws + (size_t)2*NBLK*8;    // NBLK*8 u32
  size_t slotBytes = (size_t)(2*NBLK*8 + NBLK*8) * sizeof(unsigned);
  hipMemsetAsync(d_ws, 0, slotBytes, stream);                 // reset arrival tags every call

  size_t smem = SMEM_BYTES;                                   // ~94 KB: slice fits WGP LDS (320 KB)
  hipFuncSetAttribute((const void*)rb_sim_kernel,
                      hipFuncAttributeMaxDynamicSharedMemorySize, (int)smem);
  rb_sim_kernel<<<NBLK, TPB, smem, stream>>>(x, mc, it, iq, iv, kn, mu, ld, ad,
                                             out, subSlots, inSlots);
}